// YoloLayer_7078106104485
// MI455X (gfx1250) — compile-verified
//
#include <hip/hip_runtime.h>
#include <hip/hip_bf16.h>
#include <stdint.h>

// YOLO layer: B=32, A=3, C=80, H=W=76, stride-8 anchors {1.5,2.0},{2.375,4.5},{5.0,3.5}
#define YB 32
#define YA 3
#define YC 80
#define YH 76
#define YW 76
#define YHW (YH * YW)          // 5776
#define YN  (YA * YHW)         // 17328
#define STAGE_F4 ((85 * YW) / 4)  // 1615 float4 per (b,a,h) tile
#define CONF_F4  ((YW * YC) / 4)  // 1520 float4 of conf output per tile

// ---- gfx1250 async global->LDS availability probe (compile-time) ----
#if defined(__has_builtin)
#  if __has_builtin(__builtin_amdgcn_global_load_async_to_lds_b128)
#    define USE_ASYNC_LDS 1
#  endif
#endif
#ifndef USE_ASYNC_LDS
#  define USE_ASYNC_LDS 0
#endif

#if USE_ASYNC_LDS
// Builtin signature (from hipcc diagnostic): first param is
// 'int __vector(4) __device__ *' i.e. addrspace(1) pointer to int4.
typedef int v4i __attribute__((ext_vector_type(4)));
typedef __attribute__((address_space(1))) v4i* gv4i_ptr;
typedef __attribute__((address_space(3))) v4i* lv4i_ptr;
#endif

__device__ __forceinline__ float fast_exp(float x) {
    // exp(x) = 2^(x * log2(e)) -> single v_exp_f32
    return __builtin_amdgcn_exp2f(x * 1.44269504088896340736f);
}
__device__ __forceinline__ float sigm(float x) {
    // 1/(1+e^-x) -> v_exp_f32 + v_add + v_rcp_f32
    return __builtin_amdgcn_rcpf(1.0f + fast_exp(-x));
}

// One block handles one (b, a, h) row: stages 85 channels x 76 cols into LDS,
// then writes 76 box float4s + 76*80 conf floats, all coalesced 16B stores.
__global__ __launch_bounds__(256) void yolo_decode_kernel(
        const float* __restrict__ in,   // [B, A*85, H, W]
        float* __restrict__ boxes,      // [B, N, 1, 4]
        float* __restrict__ confs) {    // [B, N, C]
    __shared__ float lds[85 * YW];      // [c][w]
    __shared__ float det[YW];           // sigmoid(objectness) per w

    const int tid = (int)threadIdx.x;
    const int blk = (int)blockIdx.x;    // 0 .. B*A*H-1
    const int h  = blk % YH;
    const int ba = blk / YH;
    const int a  = ba % YA;
    const int b  = ba / YA;

    // float offset of (channel c=0, w=0) for this (b,a,h)
    const size_t base = (size_t)(b * (YA * 85) + a * 85) * YHW + (size_t)h * YW;

    // ---- Stage 85x76 tile into LDS (row c is 76 contiguous floats) ----
#if USE_ASYNC_LDS
    for (int i = tid; i < STAGE_F4; i += 256) {
        const int c = i / 19;           // 19 float4 per 76-float row
        const int q = i - c * 19;
        const size_t g = base + (size_t)c * YHW + (size_t)(q * 4);
        __builtin_amdgcn_global_load_async_to_lds_b128(
            (gv4i_ptr)(in + g), (lv4i_ptr)&lds[4 * i], 0, 0);
    }
#  if __has_builtin(__builtin_amdgcn_s_wait_asynccnt)
    __builtin_amdgcn_s_wait_asynccnt(0);
#  else
    asm volatile("s_wait_asynccnt 0" ::: "memory");
#  endif
    __syncthreads();
#else
    for (int i = tid; i < STAGE_F4; i += 256) {
        const int c = i / 19;
        const int q = i - c * 19;
        const size_t g = base + (size_t)c * YHW + (size_t)(q * 4);
        const float4 v = *reinterpret_cast<const float4*>(in + g);
        *reinterpret_cast<float4*>(&lds[4 * i]) = v;   // lds[c*76 + q*4] == lds[4*i]
    }
    __syncthreads();
#endif

    // ---- Boxes + objectness (one cell per lane, 76 active lanes) ----
    if (tid < YW) {
        const int w = tid;
        const float sx = sigm(lds[0 * YW + w]);        // SCALE_X_Y == 1.0
        const float sy = sigm(lds[1 * YW + w]);
        const float ew = fast_exp(lds[2 * YW + w]);
        const float eh = fast_exp(lds[3 * YW + w]);
        det[w] = sigm(lds[4 * YW + w]);

        const float inv = 1.0f / 76.0f;
        const float aw = (a == 0) ? 1.5f : (a == 1) ? 2.375f : 5.0f;
        const float ah = (a == 0) ? 2.0f : (a == 1) ? 4.5f   : 3.5f;

        const float bx = (sx + (float)w) * inv;
        const float by = (sy + (float)h) * inv;
        const float bw = ew * aw * inv;
        const float bh = eh * ah * inv;

        float4 o4;
        o4.x = bx - 0.5f * bw;
        o4.y = by - 0.5f * bh;
        o4.z = o4.x + bw;
        o4.w = o4.y + bh;

        const size_t n = (size_t)a * YHW + (size_t)h * YW + (size_t)w;
        *reinterpret_cast<float4*>(boxes + ((size_t)b * YN + n) * 4) = o4;
    }
    __syncthreads();

    // ---- Class confidences: contiguous [w][c] region, float4 stores ----
    float4* confRow = reinterpret_cast<float4*>(
        confs + ((size_t)b * YN + (size_t)a * YHW + (size_t)h * YW) * YC);
    for (int i = tid; i < CONF_F4; i += 256) {
        const int w  = i / 20;          // 20 float4 of classes per cell
        const int c0 = (i - w * 20) * 4;
        const float d = det[w];
        const float* p = &lds[(5 + c0) * YW + w];
        float4 v;
        v.x = sigm(p[0 * YW]) * d;
        v.y = sigm(p[1 * YW]) * d;
        v.z = sigm(p[2 * YW]) * d;
        v.w = sigm(p[3 * YW]) * d;
        confRow[i] = v;
    }
}

extern "C" void kernel_launch(void* const* d_in, const int* in_sizes, int n_in,
                              void* d_out, int out_size, void* d_ws, size_t ws_size,
                              hipStream_t stream) {
    (void)in_sizes; (void)n_in; (void)out_size; (void)d_ws; (void)ws_size;
    const float* in = (const float*)d_in[0];
    float* boxes = (float*)d_out;                              // [B, N, 1, 4]
    float* confs = boxes + (size_t)YB * YN * 4;                // [B, N, C]
    const dim3 grid(YB * YA * YH);                             // 7296 blocks
    yolo_decode_kernel<<<grid, 256, 0, stream>>>(in, boxes, confs);
}